// GridEmbedding_64046552318105
// MI455X (gfx1250) — compile-verified
//
#include <hip/hip_runtime.h>
#include <stdint.h>

// ---------------------------------------------------------------------------
// GridEmbedding for MI455X (gfx1250, wave32).
//
// out[t,h] = emb_table[ids[t]][h] + mask[t]*((r[t]+1)*row_vec[h] + (c[t]+1)*col_vec[h])
//
// Expressed as D = A(16x4) x B(4x16) + C(16x16) per tile using
// V_WMMA_F32_16X16X4_F32.  Bandwidth-bound gather: ~512 MB moved, ~22 us
// at 23.3 TB/s; the WMMA does the rank-2 positional update for free.
// ---------------------------------------------------------------------------

typedef __attribute__((ext_vector_type(2))) float v2f;
typedef __attribute__((ext_vector_type(8))) float v8f;

#define HIDDEN      2048
#define NTOK        (4 * 8192)        // B*S = 32768 tokens
#define GROUPS      (NTOK / 16)       // 2048 token groups of 16
#define COL_TILES   (HIDDEN / 16)     // 128 column tiles of 16
#define CHUNKS      8                 // column chunks per group
#define TILES_PER_WAVE (COL_TILES / CHUNKS)   // 16 tiles per wave
#define WAVES_PER_BLOCK 8             // 256 threads = 8 wave32s on a WGP

__global__ __launch_bounds__(256)
void grid_embed_wmma_kernel(const int*     __restrict__ input_ids,   // [NTOK]
                            const int*     __restrict__ row_idx,     // [NTOK]
                            const int*     __restrict__ col_idx,     // [NTOK]
                            const uint8_t* __restrict__ grid_mask,   // [NTOK] (jnp.bool_, 1 byte)
                            const float*   __restrict__ emb_table,   // [VOCAB, HIDDEN]
                            const float*   __restrict__ row_vec,     // [HIDDEN]
                            const float*   __restrict__ col_vec,     // [HIDDEN]
                            float*         __restrict__ out)         // [NTOK, HIDDEN]
{
    const int lane = threadIdx.x & 31;
    const int li   = lane & 15;        // column within tile / row M for A
    const int half = lane >> 4;        // 0: lanes 0-15, 1: lanes 16-31

    const int unit  = blockIdx.x * WAVES_PER_BLOCK + (threadIdx.x >> 5);
    const int group = unit >> 3;                 // / CHUNKS
    const int chunk = unit & (CHUNKS - 1);
    const int t0      = group * 16;              // first token of the group
    const int h_start = chunk * (TILES_PER_WAVE * 16);

    // ------------------------------------------------------------------
    // A operand (16x4 fp32, 2 VGPRs/lane):
    //   lanes 0-15 : VGPR0 = K=0 = mask*(r+1), VGPR1 = K=1 = mask*(c+1)
    //   lanes 16-31: K=2,3 -> forced to zero (so B's K=2,3 rows are dont-care)
    // Loads are done unconditionally (addresses valid for all lanes) and
    // masked with selects -> no divergence, EXEC stays all-ones for WMMA.
    // ------------------------------------------------------------------
    const int   ta = t0 + li;
    const float m  = grid_mask[ta] ? 1.0f : 0.0f;
    const float av = m * (float)(row_idx[ta] + 1);
    const float bv = m * (float)(col_idx[ta] + 1);
    v2f A;
    A.x = half ? 0.0f : av;
    A.y = half ? 0.0f : bv;

    // ------------------------------------------------------------------
    // Hoist the gathered row bases: VGPR j of C/D holds token (t0+j) in
    // lanes 0-15 and token (t0+j+8) in lanes 16-31.
    // ------------------------------------------------------------------
    const float* crow[8];
    float*       orow[8];
#pragma unroll
    for (int j = 0; j < 8; ++j) {
        const int tj = t0 + j + (half << 3);
        crow[j] = emb_table + (size_t)input_ids[tj] * HIDDEN;
        orow[j] = out + (size_t)tj * HIDDEN;
    }

    // ------------------------------------------------------------------
    // 16 column tiles: gather C, one v_wmma_f32_16x16x4_f32, scatter D.
    // row_vec/col_vec (8 KB each) stay hot in cache.
    // ------------------------------------------------------------------
#pragma unroll 2
    for (int it = 0; it < TILES_PER_WAVE; ++it) {
        const int h = h_start + it * 16 + li;

        // B operand (4x16 fp32, 2 VGPRs/lane):
        //   lanes 0-15 : VGPR0 = row K=0 = row_vec, VGPR1 = row K=1 = col_vec
        //   lanes 16-31: rows K=2,3 = 0 (multiplied by A's zeroed K=2,3)
        const float rv = row_vec[h];
        const float cv = col_vec[h];
        v2f Bm;
        Bm.x = half ? 0.0f : rv;
        Bm.y = half ? 0.0f : cv;

        // C: gathered 16x16 token-embedding tile (8 coalesced b32 loads,
        // each covering two contiguous 64B row segments).
        v8f C;
#pragma unroll
        for (int j = 0; j < 8; ++j) C[j] = crow[j][h];

        // D = A x B + C  (8 args: neg_a, A, neg_b, B, c_mod, C, reuse_a, reuse_b)
        v8f D = __builtin_amdgcn_wmma_f32_16x16x4_f32(
            false, A, false, Bm, (short)0, C, false, false);

#pragma unroll
        for (int j = 0; j < 8; ++j) orow[j][h] = D[j];
    }
}

extern "C" void kernel_launch(void* const* d_in, const int* in_sizes, int n_in,
                              void* d_out, int out_size, void* d_ws, size_t ws_size,
                              hipStream_t stream) {
    (void)in_sizes; (void)n_in; (void)out_size; (void)d_ws; (void)ws_size;

    const int*     input_ids = (const int*)d_in[0];
    const int*     row_idx   = (const int*)d_in[1];
    const int*     col_idx   = (const int*)d_in[2];
    const uint8_t* grid_mask = (const uint8_t*)d_in[3];   // jnp.bool_ -> 1 byte/elem
    const float*   emb_table = (const float*)d_in[4];
    const float*   row_vec   = (const float*)d_in[5];     // [1, HIDDEN] -> flat
    const float*   col_vec   = (const float*)d_in[6];
    float*         out       = (float*)d_out;

    const int total_units = GROUPS * CHUNKS;              // 16384 wave-units
    const int blocks      = total_units / WAVES_PER_BLOCK; // 2048 blocks of 256 thr

    grid_embed_wmma_kernel<<<blocks, 256, 0, stream>>>(
        input_ids, row_idx, col_idx, grid_mask,
        emb_table, row_vec, col_vec, out);
}